// RHT_GC_75350906241180
// MI455X (gfx1250) — compile-verified
//
#include <hip/hip_runtime.h>

// ---------------------------------------------------------------------------
// Problem constants (from reference): N=64, CIN=64, T=128, V=25, INTER=64,
// COUT=256, L=3, S=3.  NTV = 64*128*25 = 204800, TV = 3200.
// ---------------------------------------------------------------------------
#define KN     64
#define KT     128
#define KV     25
#define KTV    3200
#define KNTV   204800
#define KEPS   1e-5f

typedef __attribute__((ext_vector_type(2))) float v2f;
typedef __attribute__((ext_vector_type(8))) float v8f;
typedef __attribute__((ext_vector_type(4))) unsigned int u32x4;
typedef __attribute__((ext_vector_type(8))) unsigned int u32x8;

__device__ __forceinline__ v8f wmma_f32(v2f a, v2f b, v8f c) {
  // D(16x16) = A(16x4) * B(4x16) + C  — fp32 WMMA (CDNA5, wave32)
  return __builtin_amdgcn_wmma_f32_16x16x4_f32(
      /*neg_a=*/false, a, /*neg_b=*/false, b,
      /*c_mod=*/(short)0, c, /*reuse_a=*/false, /*reuse_b=*/false);
}

// ---------------------------------------------------------------------------
// Tensor Data Mover: DMA a 2D tile of fp32 (tile0 x tile1 elements, rows
// separated by `stride` elements in global memory) into LDS, with optional
// LDS row padding (pad_int: pads every 2^(pad_int+1) dwords, pad_amt+1 dwords
// of padding).  D# per CDNA5 ISA; issued from one lane
// (EXEC is ignored by TDM), completion via TENSORcnt.
// ---------------------------------------------------------------------------
__device__ __forceinline__ void tdm_load_2d(const float* gaddr, float* lds_ptr,
                                            unsigned tile0, unsigned tile1,
                                            unsigned td0, unsigned td1,
                                            unsigned stride, unsigned pad_int,
                                            unsigned pad_amt, unsigned pad_en) {
  unsigned long long ga = (unsigned long long)gaddr;
  unsigned lds = (unsigned)(unsigned long long)lds_ptr;  // low 32 = LDS offset
  u32x4 g0;
  g0[0] = 1u;                                            // count=1 (valid)
  g0[1] = lds;                                           // lds_addr
  g0[2] = (unsigned)(ga & 0xffffffffu);                  // global_addr[31:0]
  g0[3] = (unsigned)((ga >> 32) & 0x01ffffffu) | (2u << 30);  // type=2
  u32x8 g1;
  g1[0] = (2u << 16) | (pad_en << 20) | (pad_int << 22) | (pad_amt << 25);
  g1[1] = (td0 & 0xffffu) << 16;                         // tensor_dim0 lo16
  g1[2] = (td0 >> 16) | ((td1 & 0xffffu) << 16);         // td0 hi | td1 lo
  g1[3] = (td1 >> 16) | ((tile0 & 0xffffu) << 16);       // td1 hi | tile_dim0
  g1[4] = tile1 & 0xffffu;                               // tile_dim1 (dim2=0)
  g1[5] = stride;                                        // dim0_stride[31:0]
  g1[6] = 0u;                                            // stride hi, dim1_stride=0
  g1[7] = 0u;
  asm volatile("tensor_load_to_lds %0, %1" ::"s"(g0), "s"(g1) : "memory");
}

// ---------------------------------------------------------------------------
// Zero fill (grid-stride)
// ---------------------------------------------------------------------------
__global__ void k_zero(float* __restrict__ p, long n) {
  long i = (long)blockIdx.x * blockDim.x + threadIdx.x;
  long stride = (long)gridDim.x * blockDim.x;
  for (; i < n; i += stride) p[i] = 0.f;
}

// ---------------------------------------------------------------------------
// Per-channel sum (+ optional sum of squares) over the (N,T,V) axes.
// One block per (n,c) row of KTV contiguous floats.  X layout: [N, C, T*V].
// ---------------------------------------------------------------------------
__global__ void k_colstats(const float* __restrict__ X, int C,
                           float* __restrict__ sums, float* __restrict__ sumsq) {
  int b = blockIdx.x;
  int c = b % C;
  const float* row = X + (size_t)b * KTV;
  float s = 0.f, s2 = 0.f;
  for (int i = threadIdx.x; i < KTV; i += blockDim.x) {
    float v = row[i];
    s += v;
    s2 += v * v;
  }
  for (int off = 16; off > 0; off >>= 1) {
    s  += __shfl_down(s, off, 32);
    s2 += __shfl_down(s2, off, 32);
  }
  __shared__ float ps[8], ps2[8];
  int wave = threadIdx.x >> 5, lane = threadIdx.x & 31;
  if (lane == 0) { ps[wave] = s; ps2[wave] = s2; }
  __syncthreads();
  if (threadIdx.x == 0) {
    int nw = blockDim.x >> 5;
    float t = 0.f, t2 = 0.f;
    for (int w = 0; w < nw; ++w) { t += ps[w]; t2 += ps2[w]; }
    atomicAdd(&sums[c], t);
    if (sumsq) atomicAdd(&sumsq[c], t2);
  }
}

// ---------------------------------------------------------------------------
// 64x64 channel gram:  G += sum_m X[c,m] * X[c',m]   (per n, over T*V)
// The A-fragment of channel-group g equals the B-fragment of group g, so
// 5 LDS fragment loads feed a row of 4 WMMA tiles per wave.
// grid = (5 chunks of 640 cols, N), block = 128 (4 waves).
// X tile staged via TDM with pad 128->132 dwords: fragment gathers (lane
// stride 132 ≡ 4 mod 64 banks) are bank-conflict free across all 32 lanes.
// ---------------------------------------------------------------------------
#define GPAD 132
__global__ void k_gram64(const float* __restrict__ X, float* __restrict__ G) {
  __shared__ float sX[64 * GPAD];
  int n = blockIdx.y;
  int c0 = blockIdx.x * 640;  // 5 chunks of 128 columns
  const float* base = X + (size_t)n * 64 * KTV + c0;
  int tid = threadIdx.x;
  int wave = tid >> 5, lane = tid & 31;
  int lr = lane & 15, hi = lane >> 4;
  v8f acc[4] = {};
  for (int ch = 0; ch < 5; ++ch) {
    __syncthreads();
    if (tid == 0) {
      tdm_load_2d(base + ch * 128, sX, 128, 64, KTV, 64, KTV, 6, 3, 1);
      __builtin_amdgcn_s_wait_tensorcnt(0);
    }
    __syncthreads();
    for (int mm = 0; mm < 128; mm += 4) {
      int off = mm + 2 * hi;
      v2f fa;
      fa.x = sX[(wave * 16 + lr) * GPAD + off];
      fa.y = sX[(wave * 16 + lr) * GPAD + off + 1];
      v2f fb[4];
#pragma unroll
      for (int g = 0; g < 4; ++g) {
        fb[g].x = sX[(g * 16 + lr) * GPAD + off];
        fb[g].y = sX[(g * 16 + lr) * GPAD + off + 1];
      }
#pragma unroll
      for (int j = 0; j < 4; ++j) acc[j] = wmma_f32(fa, fb[j], acc[j]);
    }
  }
#pragma unroll
  for (int j = 0; j < 4; ++j) {
#pragma unroll
    for (int r = 0; r < 8; ++r) {
      int row = wave * 16 + r + 8 * hi;
      int col = j * 16 + lr;
      atomicAdd(&G[row * 64 + col], acc[j][r]);
    }
  }
}

// ---------------------------------------------------------------------------
// Analytic BN fold for y = W h (+b):  mean_y = W·mean_h (+b),
// var_y[o] = W_o^T Cov(h) W_o.  Bias cancels out of the fused epilogue:
//   BN(Wh+b) = scale[o]*dot + cshift[o],
//   scale = g*rsqrt(var+eps), cshift = beta - scale*(W·mean_h).
// ---------------------------------------------------------------------------
__global__ void k_fold(const float* __restrict__ W, const float* __restrict__ g,
                       const float* __restrict__ beta,
                       const float* __restrict__ msum,
                       const float* __restrict__ gram, int O, float invCnt,
                       float* __restrict__ scale, float* __restrict__ cshift) {
  __shared__ float sMean[64];
  __shared__ float sCov[64 * 64];
  int tid = threadIdx.x;
  if (tid < 64) sMean[tid] = msum[tid] * invCnt;
  __syncthreads();
  for (int idx = tid; idx < 64 * 64; idx += blockDim.x) {
    int c = idx >> 6, cp = idx & 63;
    sCov[idx] = gram[idx] * invCnt - sMean[c] * sMean[cp];
  }
  __syncthreads();
  for (int o = tid; o < O; o += blockDim.x) {
    const float* wr = W + o * 64;
    float mdot = 0.f, var = 0.f;
    for (int c = 0; c < 64; ++c) {
      float wc = wr[c];
      mdot += wc * sMean[c];
      float rs = 0.f;
      for (int cp = 0; cp < 64; ++cp) rs += wr[cp] * sCov[c * 64 + cp];
      var += wc * rs;
    }
    float sc = g[o] * rsqrtf(var + KEPS);
    scale[o] = sc;
    cshift[o] = beta[o] - sc * mdot;
  }
}

// ---------------------------------------------------------------------------
// Adjacency mix over joints:  Z[n,c,t,v] = sum_u XD[n,c,t,u] * PA[v,u]
// One block per (n,c); the 3200-float row is staged in LDS via TDM.
// ---------------------------------------------------------------------------
__global__ void k_adj(const float* __restrict__ XD, const float* __restrict__ PA,
                      float* __restrict__ Z) {
  __shared__ float sPA[KV * KV];
  __shared__ float sRow[KTV];
  size_t base = (size_t)blockIdx.x * KTV;
  if (threadIdx.x == 0) {
    tdm_load_2d(XD + base, sRow, KTV, 1, KTV, 1, KTV, 0, 0, 0);
  }
  for (int i = threadIdx.x; i < KV * KV; i += blockDim.x) sPA[i] = PA[i];
  if (threadIdx.x == 0) __builtin_amdgcn_s_wait_tensorcnt(0);
  __syncthreads();
  for (int m = threadIdx.x; m < KTV; m += blockDim.x) {
    int t = m / KV, v = m - t * KV;
    const float* xr = &sRow[t * KV];
    const float* pr = &sPA[v * KV];
    float s = 0.f;
#pragma unroll
    for (int u = 0; u < KV; ++u) s += xr[u] * pr[u];
    Z[base + m] = s;
  }
}

// ---------------------------------------------------------------------------
// Fused conv1x1 (K=64) with folded-BN epilogue:
//   Out[n, chOff+o, m] (+)= relu?( scale[o]*sum_c W[o,c]*X[n,c,m] + cshift[o] )
// Block tile: 64 (o) x 128 (m), 8 waves.  W and X tiles DMA'd into LDS by the
// TDM with padded rows (W: 64->65, X: 128->132) for conflict-free reads.
// WMMA f32 16x16x4, 16 K-steps, 4 accumulator tiles per wave.
// grid = (25 m-tiles, N).
// ---------------------------------------------------------------------------
template <bool RELU, bool ACCUM>
__global__ void k_conv(const float* __restrict__ Xin, const float* __restrict__ Wp,
                       const float* __restrict__ scale,
                       const float* __restrict__ cshift, float* __restrict__ Out,
                       int chOff, int coutTot) {
  __shared__ float sW[64 * 65];
  __shared__ float sX[64 * 132];
  int n = blockIdx.y;
  int m0 = blockIdx.x * 128;
  int tid = threadIdx.x;
  const float* xb = Xin + (size_t)n * 64 * KTV + m0;
  if (tid == 0) {
    tdm_load_2d(Wp, sW, 4096, 1, 4096, 1, 4096, 5, 0, 1);  // rows 64 -> 65
    tdm_load_2d(xb, sX, 128, 64, KTV, 64, KTV, 6, 3, 1);   // rows 128 -> 132
    __builtin_amdgcn_s_wait_tensorcnt(0);
  }
  __syncthreads();
  int wave = tid >> 5, lane = tid & 31;
  int o_sub = (wave & 3) * 16;
  int m_sub = (wave >> 2) * 64;
  int lr = lane & 15, hi = lane >> 4;
  v8f acc[4] = {};
  for (int kk = 0; kk < 64; kk += 4) {
    v2f a;
    const float* ap = &sW[(o_sub + lr) * 65 + kk + 2 * hi];
    a.x = ap[0];
    a.y = ap[1];
#pragma unroll
    for (int ct = 0; ct < 4; ++ct) {
      int col = m_sub + ct * 16 + lr;
      v2f b;
      b.x = sX[(kk + 2 * hi) * 132 + col];
      b.y = sX[(kk + 2 * hi + 1) * 132 + col];
      acc[ct] = wmma_f32(a, b, acc[ct]);
    }
  }
#pragma unroll
  for (int ct = 0; ct < 4; ++ct) {
#pragma unroll
    for (int r = 0; r < 8; ++r) {
      int o = o_sub + r + 8 * hi;
      int col = m_sub + ct * 16 + lr;
      float val = scale[o] * acc[ct][r] + cshift[o];
      if (RELU) val = fmaxf(val, 0.f);
      size_t oi = ((size_t)n * coutTot + chOff + o) * KTV + m0 + col;
      if (ACCUM)
        Out[oi] += val;
      else
        Out[oi] = val;
    }
  }
}

// ---------------------------------------------------------------------------
// Final fused kernel:
//   out = relu( BN_fin(ACC) + folded-BN(down_w · x) )
// grid = (25 m-tiles, 4 o-groups of 64, N).  ACC lives in d_out; each element
// is read exactly once by its owning lane before being overwritten.
// ---------------------------------------------------------------------------
__global__ void k_final(const float* __restrict__ Xin, const float* __restrict__ DW,
                        const float* __restrict__ scale,
                        const float* __restrict__ cshift,
                        const float* __restrict__ ACC,
                        const float* __restrict__ fsum, const float* __restrict__ fsq,
                        const float* __restrict__ gfin, const float* __restrict__ bfin,
                        float* __restrict__ Out) {
  __shared__ float sW[64 * 65];
  __shared__ float sX[64 * 132];
  int n = blockIdx.z;
  int og = blockIdx.y;
  int m0 = blockIdx.x * 128;
  int tid = threadIdx.x;
  const float* Wp = DW + (size_t)og * 64 * 64;
  const float* xb = Xin + (size_t)n * 64 * KTV + m0;
  if (tid == 0) {
    tdm_load_2d(Wp, sW, 4096, 1, 4096, 1, 4096, 5, 0, 1);
    tdm_load_2d(xb, sX, 128, 64, KTV, 64, KTV, 6, 3, 1);
    __builtin_amdgcn_s_wait_tensorcnt(0);
  }
  __syncthreads();
  int wave = tid >> 5, lane = tid & 31;
  int o_sub = (wave & 3) * 16;
  int m_sub = (wave >> 2) * 64;
  int lr = lane & 15, hi = lane >> 4;
  v8f acc[4] = {};
  for (int kk = 0; kk < 64; kk += 4) {
    v2f a;
    const float* ap = &sW[(o_sub + lr) * 65 + kk + 2 * hi];
    a.x = ap[0];
    a.y = ap[1];
#pragma unroll
    for (int ct = 0; ct < 4; ++ct) {
      int col = m_sub + ct * 16 + lr;
      v2f b;
      b.x = sX[(kk + 2 * hi) * 132 + col];
      b.y = sX[(kk + 2 * hi + 1) * 132 + col];
      acc[ct] = wmma_f32(a, b, acc[ct]);
    }
  }
  const float invC = 1.f / (float)KNTV;
#pragma unroll
  for (int ct = 0; ct < 4; ++ct) {
#pragma unroll
    for (int r = 0; r < 8; ++r) {
      int o_l = o_sub + r + 8 * hi;
      int oc = og * 64 + o_l;
      int col = m_sub + ct * 16 + lr;
      size_t idx = ((size_t)n * 256 + oc) * KTV + m0 + col;
      float dval = scale[oc] * acc[ct][r] + cshift[oc];
      float a = ACC[idx];
      float mean = fsum[oc] * invC;
      float var = fsq[oc] * invC - mean * mean;
      float nrm = (a - mean) * rsqrtf(var + KEPS) * gfin[oc] + bfin[oc];
      Out[idx] = fmaxf(nrm + dval, 0.f);
    }
  }
}

// ---------------------------------------------------------------------------
// Host-side orchestration
// ---------------------------------------------------------------------------
extern "C" void kernel_launch(void* const* d_in, const int* in_sizes, int n_in,
                              void* d_out, int out_size, void* d_ws, size_t ws_size,
                              hipStream_t stream) {
  (void)in_sizes; (void)n_in; (void)out_size; (void)ws_size;
  const float* x        = (const float*)d_in[0];
  const float* PA       = (const float*)d_in[1];
  const float* wd       = (const float*)d_in[2];
  // d_in[3] = bd : cancels out of folded BN
  const float* gd       = (const float*)d_in[4];
  const float* betad    = (const float*)d_in[5];
  const float* wsm      = (const float*)d_in[6];
  // d_in[7] = bs : cancels out of folded BN
  const float* gs       = (const float*)d_in[8];
  const float* betas    = (const float*)d_in[9];
  const float* down_w   = (const float*)d_in[10];
  // d_in[11] = down_b : cancels out of folded BN
  const float* down_g   = (const float*)d_in[12];
  const float* down_beta= (const float*)d_in[13];
  const float* gfin     = (const float*)d_in[14];
  const float* bfin     = (const float*)d_in[15];
  float* out = (float*)d_out;

  // Workspace layout (floats).  ACC (N*256*TV) lives in d_out.
  float* XD  = (float*)d_ws;          // 13,107,200
  float* Z   = XD + 13107200;         // 13,107,200
  float* GRX = Z + 13107200;          // 4096  gram(x)
  float* MSX = GRX + 4096;            // 64    sum(x)
  float* GRT = MSX + 64;              // 4096  gram(tmp)
  float* MST = GRT + 4096;            // 64    sum(tmp)
  float* SCD = MST + 64;              // 64    folded scale (conv_d)
  float* SHD = SCD + 64;              // 64
  float* SCS = SHD + 64;              // 64    folded scale (branch conv)
  float* SHS = SCS + 64;              // 64
  float* SCW = SHS + 64;              // 256   folded scale (down conv)
  float* SHW = SCW + 256;             // 256
  float* FSU = SHW + 256;             // 256   final sums
  float* FSQ = FSU + 256;             // 256   final sumsq
  float* ACC = out;

  const float invNTV = 1.f / (float)KNTV;
  const long statsLen = 4096 + 64 + 4096 + 64 + 4 * 64 + 4 * 256;

  k_zero<<<2048, 256, 0, stream>>>(ACC, (long)KN * 256 * KTV);
  k_zero<<<64, 256, 0, stream>>>(GRX, statsLen);

  // Input statistics: mean + 64x64 gram (shared by all conv_d folds + down)
  k_colstats<<<KN * 64, 128, 0, stream>>>(x, 64, MSX, nullptr);
  k_gram64<<<dim3(5, KN), 128, 0, stream>>>(x, GRX);
  k_fold<<<1, 256, 0, stream>>>(down_w, down_g, down_beta, MSX, GRX, 256,
                                invNTV, SCW, SHW);

  for (int l = 0; l < 3; ++l) {
    const float* wdl = wd + l * 4096;
    // xd = relu(BN(conv_d(x))) with BN folded analytically
    k_fold<<<1, 256, 0, stream>>>(wdl, gd + l * 64, betad + l * 64, MSX, GRX,
                                  64, invNTV, SCD, SHD);
    k_conv<true, false><<<dim3(25, KN), 256, 0, stream>>>(x, wdl, SCD, SHD, XD,
                                                          0, 64);
    // Statistics of xd (for edge-branch BN fold)
    k_zero<<<32, 256, 0, stream>>>(GRT, 4096 + 64);
    k_colstats<<<KN * 64, 128, 0, stream>>>(XD, 64, MST, nullptr);
    k_gram64<<<dim3(5, KN), 128, 0, stream>>>(XD, GRT);

    // Edge branch: BN(conv(xd, ws[l,S-1])) -> ACC channels [192:256)
    const float* wse = wsm + (l * 3 + 2) * 4096;
    k_fold<<<1, 256, 0, stream>>>(wse, gs + (l * 3 + 2) * 64,
                                  betas + (l * 3 + 2) * 64, MST, GRT, 64,
                                  invNTV, SCS, SHS);
    k_conv<false, true><<<dim3(25, KN), 256, 0, stream>>>(XD, wse, SCS, SHS,
                                                          ACC, 192, 256);

    // Subset branches: z = adj(xd), BN(conv(z)) -> ACC channels [64j:64j+64)
    for (int j = 0; j < 3; ++j) {
      k_adj<<<KN * 64, 256, 0, stream>>>(XD, PA + (l * 3 + j) * 625, Z);
      k_zero<<<32, 256, 0, stream>>>(GRT, 4096 + 64);
      k_colstats<<<KN * 64, 128, 0, stream>>>(Z, 64, MST, nullptr);
      k_gram64<<<dim3(5, KN), 128, 0, stream>>>(Z, GRT);
      const float* wsj = wsm + (l * 3 + j) * 4096;
      k_fold<<<1, 256, 0, stream>>>(wsj, gs + (l * 3 + j) * 64,
                                    betas + (l * 3 + j) * 64, MST, GRT, 64,
                                    invNTV, SCS, SHS);
      k_conv<false, true><<<dim3(25, KN), 256, 0, stream>>>(Z, wsj, SCS, SHS,
                                                            ACC, j * 64, 256);
    }
  }

  // Final BN stats over ACC, then fused down-conv + final BN + relu
  k_colstats<<<KN * 256, 128, 0, stream>>>(ACC, 256, FSU, FSQ);
  k_final<<<dim3(25, 4, KN), 256, 0, stream>>>(x, down_w, SCW, SHW, ACC, FSU,
                                               FSQ, gfin, bfin, out);
}